// EfficientMemoryGELU_23106924052985
// MI455X (gfx1250) — compile-verified
//
#include <hip/hip_runtime.h>
#include <math.h>

// ---------------------------------------------------------------------------
// EfficientMemoryGELU for MI455X (gfx1250).
//
// Memory-bound: 128MB in, 256MB+4B out -> ~16.5us floor at 23.3 TB/s.
//  * exact 3-level radix-select (11/11/10 bits) for kthvalue; the whole
//    128MB input stays resident in the 192MB L2 across the rescans.
//  * level-0 (HBM-cold) scan stages data with double-buffered
//    global_load_async_to_lds_b128 (ASYNCcnt pipeline: issue i+1, wait for
//    i with s_wait_asynccnt 1) and counts with ds_add_u32 LDS histograms.
//  * final fused GELU+prune pass streams its 256MB of outputs with
//    non-temporal b128 stores so the L2-resident input isn't thrashed.
// ---------------------------------------------------------------------------

typedef float fvec4 __attribute__((ext_vector_type(4)));

struct Ctl {
    unsigned prefix;   // selected key prefix so far
    unsigned rank;     // remaining 1-indexed rank within prefix
    unsigned kth_key;  // final 32-bit key
    float    kth_val;  // final threshold value
};

// Order-preserving float->uint key (no NaNs in gaussian input).
__device__ __forceinline__ unsigned f2key(float f) {
    unsigned b = __float_as_uint(f);
    return (b & 0x80000000u) ? ~b : (b | 0x80000000u);
}
__device__ __forceinline__ float key2f(unsigned k) {
    unsigned b = (k & 0x80000000u) ? (k ^ 0x80000000u) : ~k;
    return __uint_as_float(b);
}

__device__ __forceinline__ void async_copy_b128(unsigned lds_byte,
                                                const void* gaddr) {
    asm volatile("global_load_async_to_lds_b128 %0, %1, off"
                 :
                 : "v"(lds_byte), "v"((unsigned long long)gaddr)
                 : "memory");
}

// ---------------------------------------------------------------------------
// Level-0 histogram (top 11 key bits, 2048 bins), HBM-cold scan.
// Double-buffered async global->LDS staging: keep 2 copies in flight per
// wave; wait only for the older one (ASYNCcnt decrements in issue order).
// ---------------------------------------------------------------------------
__global__ void __launch_bounds__(256)
hist_l0_kernel(const float* __restrict__ x, long long n4,
               unsigned* __restrict__ hist) {
    __shared__ unsigned lh[2048];
    __shared__ float    tile[2][256 * 4];

    const int tid = threadIdx.x;
    for (int b = tid; b < 2048; b += 256) lh[b] = 0u;
    __syncthreads();

    // Low 32 bits of a generic pointer into LDS == LDS byte offset.
    unsigned lds_off[2];
    lds_off[0] = (unsigned)(unsigned long long)(const void*)&tile[0][tid * 4];
    lds_off[1] = (unsigned)(unsigned long long)(const void*)&tile[1][tid * 4];

    const long long stride = (long long)gridDim.x * 256;
    long long i = (long long)blockIdx.x * 256 + tid;
    bool valid = (i < n4);
    int cur = 0;

    if (valid) async_copy_b128(lds_off[0], x + i * 4);   // prologue

    while (valid) {
        const long long inext = i + stride;
        const bool vnext = (inext < n4);
        if (vnext) {
            async_copy_b128(lds_off[cur ^ 1], x + inext * 4);
            asm volatile("s_wait_asynccnt 0x1" ::: "memory"); // older copy done
        } else {
            asm volatile("s_wait_asynccnt 0x0" ::: "memory");
        }
        const float* tv = &tile[cur][tid * 4];
        #pragma unroll
        for (int j = 0; j < 4; ++j) {
            unsigned key = f2key(tv[j]);
            atomicAdd(&lh[key >> 21], 1u);                // ds_add_u32
        }
        i = inext; valid = vnext; cur ^= 1;
    }
    __syncthreads();
    for (int b = tid; b < 2048; b += 256)
        if (lh[b]) atomicAdd(&hist[b], lh[b]);
}

// ---------------------------------------------------------------------------
// Level 1/2 histograms: rescan L2-resident input, count only keys matching
// the prefix chosen so far. 2x unrolled so two b128 loads are in flight.
//   level==1: match key>>21, bin=(key>>10)&0x7FF  (2048 bins)
//   level==2: match key>>10, bin= key&0x3FF       (1024 bins)
// ---------------------------------------------------------------------------
__global__ void __launch_bounds__(256)
hist_ln_kernel(const float* __restrict__ x, long long n4,
               unsigned* __restrict__ hist, const Ctl* __restrict__ ctl,
               int level) {
    __shared__ unsigned lh[2048];
    const int tid = threadIdx.x;
    const int nbins = (level == 1) ? 2048 : 1024;
    for (int b = tid; b < nbins; b += 256) lh[b] = 0u;
    __syncthreads();

    const unsigned pref = ctl->prefix;   // uniform -> scalar load
    const fvec4* xv = (const fvec4*)x;
    const long long stride = (long long)gridDim.x * 256;

    long long i = (long long)blockIdx.x * 256 + tid;
    for (; i + stride < n4; i += 2 * stride) {
        fvec4 v0 = xv[i];
        fvec4 v1 = xv[i + stride];
        #pragma unroll
        for (int j = 0; j < 4; ++j) {
            unsigned k0 = f2key(v0[j]);
            unsigned k1 = f2key(v1[j]);
            if (level == 1) {
                if ((k0 >> 21) == pref) atomicAdd(&lh[(k0 >> 10) & 0x7FFu], 1u);
                if ((k1 >> 21) == pref) atomicAdd(&lh[(k1 >> 10) & 0x7FFu], 1u);
            } else {
                if ((k0 >> 10) == pref) atomicAdd(&lh[k0 & 0x3FFu], 1u);
                if ((k1 >> 10) == pref) atomicAdd(&lh[k1 & 0x3FFu], 1u);
            }
        }
    }
    if (i < n4) {                        // remainder (empty for 33.5M input)
        fvec4 v = xv[i];
        #pragma unroll
        for (int j = 0; j < 4; ++j) {
            unsigned k = f2key(v[j]);
            if (level == 1) {
                if ((k >> 21) == pref) atomicAdd(&lh[(k >> 10) & 0x7FFu], 1u);
            } else {
                if ((k >> 10) == pref) atomicAdd(&lh[k & 0x3FFu], 1u);
            }
        }
    }
    __syncthreads();
    for (int b = tid; b < nbins; b += 256)
        if (lh[b]) atomicAdd(&hist[b], lh[b]);
}

// ---------------------------------------------------------------------------
// Select the bin holding the (1-indexed) target rank; refine prefix + rank.
// Tiny serial scan (<=2048 adds) — negligible next to the 128MB scans.
// ---------------------------------------------------------------------------
__global__ void select_kernel(const unsigned* __restrict__ hist, int nbins,
                              int bits, Ctl* __restrict__ ctl, int level,
                              long long k0, float* __restrict__ kth_out) {
    if (threadIdx.x != 0 || blockIdx.x != 0) return;
    long long k   = (level == 0) ? k0 : (long long)ctl->rank;
    unsigned pref = (level == 0) ? 0u : ctl->prefix;

    long long cum = 0;
    unsigned  sel = 0;
    for (int b = 0; b < nbins; ++b) {
        unsigned c = hist[b];
        if (cum + (long long)c >= k) { sel = (unsigned)b; k -= cum; break; }
        cum += c;
    }
    unsigned newpref = (pref << bits) | sel;
    ctl->prefix = newpref;
    ctl->rank   = (unsigned)k;
    if (level == 2) {
        ctl->kth_key = newpref;
        float kv = key2f(newpref);
        ctl->kth_val = kv;
        *kth_out = kv;               // third output of the reference tuple
    }
}

// ---------------------------------------------------------------------------
// Fused GELU (exact erf) + prune. Loads hit L2 (input resident from scans);
// both 128MB output streams use non-temporal b128 stores. 2x unrolled.
// ---------------------------------------------------------------------------
__global__ void __launch_bounds__(256)
finalize_kernel(const float* __restrict__ x, float* __restrict__ out_gelu,
                float* __restrict__ out_pruned, const Ctl* __restrict__ ctl,
                long long n4) {
    const float kth = ctl->kth_val;              // uniform scalar
    const float INV_SQRT2 = 0.70710678f;         // np.float32(1/sqrt(2))
    const fvec4* xv = (const fvec4*)x;
    fvec4* gv = (fvec4*)out_gelu;
    fvec4* pv = (fvec4*)out_pruned;

    const long long stride = (long long)gridDim.x * 256;
    long long i = (long long)blockIdx.x * 256 + threadIdx.x;
    for (; i + stride < n4; i += 2 * stride) {
        fvec4 v0 = xv[i];
        fvec4 v1 = xv[i + stride];
        fvec4 g0, p0, g1, p1;
        #pragma unroll
        for (int j = 0; j < 4; ++j) {
            float a = v0[j], b = v1[j];
            g0[j] = 0.5f * a * (1.0f + erff(a * INV_SQRT2));
            g1[j] = 0.5f * b * (1.0f + erff(b * INV_SQRT2));
            p0[j] = (a > kth) ? a : -10.0f;
            p1[j] = (b > kth) ? b : -10.0f;
        }
        __builtin_nontemporal_store(g0, gv + i);            // th:TH_STORE_NT
        __builtin_nontemporal_store(p0, pv + i);
        __builtin_nontemporal_store(g1, gv + i + stride);
        __builtin_nontemporal_store(p1, pv + i + stride);
    }
    if (i < n4) {
        fvec4 v = xv[i];
        fvec4 g, p;
        #pragma unroll
        for (int j = 0; j < 4; ++j) {
            float a = v[j];
            g[j] = 0.5f * a * (1.0f + erff(a * INV_SQRT2));
            p[j] = (a > kth) ? a : -10.0f;
        }
        __builtin_nontemporal_store(g, gv + i);
        __builtin_nontemporal_store(p, pv + i);
    }
}

// ---------------------------------------------------------------------------
// Workspace layout (20,496 bytes):
//   [0,8192)      hist level 0 (2048 u32)
//   [8192,16384)  hist level 1 (2048 u32)
//   [16384,20480) hist level 2 (1024 u32)
//   [20480,20496) Ctl
// ---------------------------------------------------------------------------
extern "C" void kernel_launch(void* const* d_in, const int* in_sizes, int n_in,
                              void* d_out, int out_size, void* d_ws,
                              size_t ws_size, hipStream_t stream) {
    (void)n_in; (void)out_size; (void)ws_size;

    const float* x = (const float*)d_in[0];
    const long long n  = (long long)in_sizes[0];   // 33,554,432
    const long long n4 = n / 4;                    // divisible (last dim 2048)

    float* out_gelu   = (float*)d_out;
    float* out_pruned = out_gelu + n;
    float* kth_out    = out_gelu + 2 * n;

    unsigned char* ws = (unsigned char*)d_ws;
    unsigned* hist0 = (unsigned*)(ws);
    unsigned* hist1 = (unsigned*)(ws + 8192);
    unsigned* hist2 = (unsigned*)(ws + 16384);
    Ctl*      ctl   = (Ctl*)(ws + 20480);

    // k = int(N * 0.9): same double-precision truncation as Python.
    const long long k = (long long)((double)n * 0.9);

    hipMemsetAsync(d_ws, 0, 20496, stream);

    const dim3 blk(256);
    const int hist_blocks = 4096;   // 1M threads, 8 float4 iters each

    hist_l0_kernel<<<hist_blocks, blk, 0, stream>>>(x, n4, hist0);
    select_kernel<<<1, 32, 0, stream>>>(hist0, 2048, 11, ctl, 0, k, kth_out);

    hist_ln_kernel<<<hist_blocks, blk, 0, stream>>>(x, n4, hist1, ctl, 1);
    select_kernel<<<1, 32, 0, stream>>>(hist1, 2048, 11, ctl, 1, k, kth_out);

    hist_ln_kernel<<<hist_blocks, blk, 0, stream>>>(x, n4, hist2, ctl, 2);
    select_kernel<<<1, 32, 0, stream>>>(hist2, 1024, 10, ctl, 2, k, kth_out);

    finalize_kernel<<<8192, blk, 0, stream>>>(x, out_gelu, out_pruned, ctl, n4);
}